// TorchGeoGraphAttentionEncoder_9637906612579
// MI455X (gfx1250) — compile-verified
//
#include <hip/hip_runtime.h>

// ---------------------------------------------------------------------------
// TorchGeo GraphAttentionEncoder for MI455X (gfx1250, wave32, WMMA).
//
// Algebraic reduction (exact in fp32): the reference aggregates V[dst] with
// segment-softmax weights that sum to 1 (max-subtraction guarantees s >= 1,
// so s + 1e-16 == s). Hence aggr[n] = mask[n] * V_viewed[n] where mask[n] is
// "node n has >= 1 incoming edge". Q/K/softmax are dead code.
//
// The torch .view(BN,H,K) of an [H,BN,K] tensor is handled by storing V in
// [H,BN,K] layout and re-reading the same flat buffer as row-major [BN,128].
//
// GEMMs use V_WMMA_F32_16X16X4_F32 (full fp32 precision; the workload is
// bandwidth-bound, not compute-bound). Each wave computes RT=5 row-tiles that
// share one set of B fragments -> 5 WMMAs per 2 B-dword loads.
// ---------------------------------------------------------------------------

typedef float v2f __attribute__((ext_vector_type(2)));
typedef float v8f __attribute__((ext_vector_type(8)));

#define NNODES 20000      // B*N
#define DM     128        // embed dim
#define NHEAD  8
#define KD     16
#define FF     512
#define NLAYER 3
#define NEDGE  320000
#define BN_EPS 1e-5f
#define RT     5          // row tiles per wave (80 rows/block, 20000/80=250)
#define KB     128        // K chunk staged in LDS

// ---------------------------------------------------------------------- utils
__global__ __launch_bounds__(256) void zero_kernel(unsigned int* p, int n) {
  int i = blockIdx.x * blockDim.x + threadIdx.x;
  if (i < n) p[i] = 0u;
}

__global__ __launch_bounds__(256) void copy_kernel(const float* __restrict__ src,
                                                   float* __restrict__ dst, int n) {
  int i = blockIdx.x * blockDim.x + threadIdx.x;
  if (i < n) dst[i] = src[i];
}

// mark nodes that receive at least one edge (idempotent plain store)
__global__ __launch_bounds__(256) void mask_kernel(const int* __restrict__ dst_idx,
                                                   int* __restrict__ mask, int ne) {
  int e = blockIdx.x * blockDim.x + threadIdx.x;
  if (e < ne) mask[dst_idx[e]] = 1;
}

// ------------------------------------------------------------- V projection
// Vbuf[h][n][k] = sum_d X[n][d] * Wv[h][d][k]   (einsum 'nd,hdk->hnk' layout)
// block = 256 threads = 8 waves; wave w == head w; 80-row tile per block.
__global__ __launch_bounds__(256)
void vproj_kernel(const float* __restrict__ X, const float* __restrict__ Wv,
                  float* __restrict__ Vbuf) {
  __shared__ float xs[RT * 16 * DM];                  // 40 KB A-tile
  const int r0 = blockIdx.x * (RT * 16);
  for (int i = threadIdx.x; i < RT * 16 * DM; i += 256) {
    int m = i >> 7, d = i & (DM - 1);
    xs[i] = X[(r0 + m) * DM + d];
  }
  __syncthreads();

  const int wave = threadIdx.x >> 5;                  // head
  const int lane = threadIdx.x & 31;
  const int mrowA = lane & 15;                        // A: lane -> M
  const int ncol  = lane & 15;                        // B: lane -> N
  const int koff  = (lane >> 4) * 2;                  // K split across halves
  const float* W  = Wv + wave * (DM * KD);            // [128 x 16] row-major

  v8f acc[RT];
  for (int t = 0; t < RT; ++t) acc[t] = (v8f){};

  for (int k = 0; k < DM; k += 4) {
    v2f b;
    b.x = W[(k + koff) * KD + ncol];
    b.y = W[(k + koff + 1) * KD + ncol];
    for (int t = 0; t < RT; ++t) {
      v2f a;
      a.x = xs[(t * 16 + mrowA) * DM + k + koff];
      a.y = xs[(t * 16 + mrowA) * DM + k + koff + 1];
      acc[t] = __builtin_amdgcn_wmma_f32_16x16x4_f32(false, a, false, b,
                                                     (short)0, acc[t], false, false);
    }
  }
  // C layout: VGPR r -> (M=r, lanes 0-15) / (M=8+r, lanes 16-31)
  const int half = lane >> 4;
  for (int t = 0; t < RT; ++t)
    for (int r = 0; r < 8; ++r) {
      int row = r0 + t * 16 + r + 8 * half;
      Vbuf[wave * (NNODES * KD) + row * KD + ncol] = acc[t][r];
    }
}

// -------------------------------------------------------- generic WMMA GEMM
// C[M x N] = epilogue( A[M x K] @ Bw[K x N] )
//   optional: per-row mask on A, bias[col], relu, residual[row*N+col]
// block = 256 threads = 8 waves covering 80 rows x 128 cols; grid.y = N/128.
// K is processed in KB=128 chunks staged in LDS.
__global__ __launch_bounds__(256)
void gemm_kernel(const float* __restrict__ A, const float* __restrict__ Bw,
                 const float* __restrict__ bias, const float* __restrict__ resid,
                 const int* __restrict__ rowmask, float* __restrict__ C,
                 int K, int N, int do_relu) {
  __shared__ float xs[RT * 16 * KB];                  // 40 KB A-chunk
  const int r0 = blockIdx.x * (RT * 16);
  const int wave = threadIdx.x >> 5;
  const int lane = threadIdx.x & 31;
  const int colbase = blockIdx.y * 128 + wave * 16;
  const int mrowA = lane & 15;
  const int ncol  = lane & 15;
  const int koff  = (lane >> 4) * 2;

  v8f acc[RT];
  for (int t = 0; t < RT; ++t) acc[t] = (v8f){};

  for (int kc = 0; kc < K; kc += KB) {
    for (int i = threadIdx.x; i < RT * 16 * KB; i += 256) {
      int m = i >> 7, d = i & (KB - 1);
      float v = A[(r0 + m) * K + kc + d];
      if (rowmask && rowmask[r0 + m] == 0) v = 0.0f;
      xs[i] = v;
    }
    __syncthreads();

    for (int k = 0; k < KB; k += 4) {
      v2f b;
      b.x = Bw[(kc + k + koff) * N + colbase + ncol];
      b.y = Bw[(kc + k + koff + 1) * N + colbase + ncol];
      for (int t = 0; t < RT; ++t) {
        v2f a;
        a.x = xs[(t * 16 + mrowA) * KB + k + koff];
        a.y = xs[(t * 16 + mrowA) * KB + k + koff + 1];
        acc[t] = __builtin_amdgcn_wmma_f32_16x16x4_f32(false, a, false, b,
                                                       (short)0, acc[t], false, false);
      }
    }
    __syncthreads();
  }

  const int half = lane >> 4;
  for (int t = 0; t < RT; ++t)
    for (int r = 0; r < 8; ++r) {
      int row = r0 + t * 16 + r + 8 * half;
      int col = colbase + ncol;
      float v = acc[t][r];
      if (bias)    v += bias[col];
      if (do_relu) v  = fmaxf(v, 0.0f);
      if (resid)   v += resid[row * N + col];
      C[row * N + col] = v;
    }
}

// --------------------------------------------------------------- batch norm
// column-wise mean/var over 20000 rows. 128 threads/block, one column each;
// full rows are read coalesced; partial sums land in 128-entry global bufs.
__global__ __launch_bounds__(128)
void bn_stats_kernel(const float* __restrict__ X, float* __restrict__ sums,
                     float* __restrict__ sqs) {
  const int d = threadIdx.x;
  const int rows_per = NNODES / gridDim.x;            // grid evenly divides
  const int rbeg = blockIdx.x * rows_per;
  const int rend = rbeg + rows_per;
  float s = 0.0f, q = 0.0f;
  for (int r = rbeg; r < rend; ++r) {
    float v = X[r * DM + d];
    s += v;
    q += v * v;
  }
  atomicAdd(&sums[d], s);
  atomicAdd(&sqs[d], q);
}

__global__ __launch_bounds__(256)
void bn_apply_kernel(const float* __restrict__ X, const float* __restrict__ sums,
                     const float* __restrict__ sqs, const float* __restrict__ w,
                     const float* __restrict__ b, float* __restrict__ Y) {
  int i = blockIdx.x * blockDim.x + threadIdx.x;
  if (i >= NNODES * DM) return;
  int d = i & (DM - 1);
  float mu  = sums[d] * (1.0f / NNODES);
  float var = sqs[d] * (1.0f / NNODES) - mu * mu;     // biased var, as jnp.var
  float rstd = rsqrtf(var + BN_EPS);
  Y[i] = (X[i] - mu) * rstd * w[d] + b[d];
}

// -------------------------------------------------------------------- driver
extern "C" void kernel_launch(void* const* d_in, const int* in_sizes, int n_in,
                              void* d_out, int out_size, void* d_ws, size_t ws_size,
                              hipStream_t stream) {
  const float* x    = (const float*)d_in[0];
  const int*   eidx = (const int*)d_in[1];
  // d_in[2]=Wq, d_in[3]=Wk : dead (softmax weights sum to 1, V uses dst)
  const float* Wv   = (const float*)d_in[4];
  const float* Wout = (const float*)d_in[5];
  const float* bn1w = (const float*)d_in[6];
  const float* bn1b = (const float*)d_in[7];
  const float* W1   = (const float*)d_in[8];
  const float* b1   = (const float*)d_in[9];
  const float* W2   = (const float*)d_in[10];
  const float* b2   = (const float*)d_in[11];
  const float* bn2w = (const float*)d_in[12];
  const float* bn2b = (const float*)d_in[13];
  float* out = (float*)d_out;

  // workspace layout
  char* ws = (char*)d_ws;
  size_t off = 0;
  auto alloc = [&](size_t bytes) {
    void* p = ws + off;
    off = (off + bytes + 255) & ~(size_t)255;
    return p;
  };
  float* hbuf = (float*)alloc((size_t)NNODES * DM * 4);
  float* htmp = (float*)alloc((size_t)NNODES * DM * 4);
  float* Vbuf = (float*)alloc((size_t)NNODES * DM * 4);
  float* Fbuf = (float*)alloc((size_t)NNODES * FF * 4);
  int*   mask = (int*)alloc((size_t)NNODES * 4);
  float* sums = (float*)alloc(DM * 4);
  float* sqs  = (float*)alloc(DM * 4);

  const int MBLOCKS = NNODES / (RT * 16);             // 250

  // h = x
  copy_kernel<<<(NNODES * DM + 255) / 256, 256, 0, stream>>>(x, hbuf, NNODES * DM);

  // incoming-edge mask (edges identical across layers)
  zero_kernel<<<(NNODES + 255) / 256, 256, 0, stream>>>((unsigned int*)mask, NNODES);
  mask_kernel<<<(NEDGE + 255) / 256, 256, 0, stream>>>(eidx + NEDGE, mask, NEDGE);

  for (int l = 0; l < NLAYER; ++l) {
    const float* Wv_l   = Wv   + (size_t)l * NHEAD * DM * KD;
    const float* Wout_l = Wout + (size_t)l * NHEAD * KD * DM;   // [128 x 128]
    const float* W1_l   = W1   + (size_t)l * DM * FF;
    const float* b1_l   = b1   + (size_t)l * FF;
    const float* W2_l   = W2   + (size_t)l * FF * DM;
    const float* b2_l   = b2   + (size_t)l * DM;

    // V projection: Vbuf in [H,NNODES,KD] layout == flat [NNODES,128] view
    vproj_kernel<<<MBLOCKS, 256, 0, stream>>>(hbuf, Wv_l, Vbuf);

    // attention output + residual: htmp = hbuf + (mask * Vview) @ WoutR
    gemm_kernel<<<dim3(MBLOCKS, 1), 256, 0, stream>>>(
        Vbuf, Wout_l, nullptr, hbuf, mask, htmp, DM, DM, 0);

    // BatchNorm1 -> hbuf
    zero_kernel<<<1, 256, 0, stream>>>((unsigned int*)sums, DM);
    zero_kernel<<<1, 256, 0, stream>>>((unsigned int*)sqs, DM);
    bn_stats_kernel<<<100, 128, 0, stream>>>(htmp, sums, sqs);
    bn_apply_kernel<<<(NNODES * DM + 255) / 256, 256, 0, stream>>>(
        htmp, sums, sqs, bn1w + l * DM, bn1b + l * DM, hbuf);

    // FFN: F = relu(h @ W1 + b1); htmp = h + F @ W2 + b2
    gemm_kernel<<<dim3(MBLOCKS, FF / 128), 256, 0, stream>>>(
        hbuf, W1_l, b1_l, nullptr, nullptr, Fbuf, DM, FF, 1);
    gemm_kernel<<<dim3(MBLOCKS, 1), 256, 0, stream>>>(
        Fbuf, W2_l, b2_l, hbuf, nullptr, htmp, FF, DM, 0);

    // BatchNorm2 -> hbuf (or d_out on final layer)
    zero_kernel<<<1, 256, 0, stream>>>((unsigned int*)sums, DM);
    zero_kernel<<<1, 256, 0, stream>>>((unsigned int*)sqs, DM);
    bn_stats_kernel<<<100, 128, 0, stream>>>(htmp, sums, sqs);
    float* dst = (l == NLAYER - 1) ? out : hbuf;
    bn_apply_kernel<<<(NNODES * DM + 255) / 256, 256, 0, stream>>>(
        htmp, sums, sqs, bn2w + l * DM, bn2b + l * DM, dst);
  }
}